// GCN_34230889349670
// MI455X (gfx1250) — compile-verified
//
#include <hip/hip_runtime.h>
#include <hip/hip_bf16.h>

// ---------------------------------------------------------------------------
// GCN (3-layer) for MI455X / gfx1250, wave32.
//
//   1. degree count (int atomics)                 -> deg[]
//   2. dinv[i] = rsqrt(deg[i]+1)  (self-loop incl.)
//   3. exclusive scan of deg -> row_ptr (CSR by dst)
//   4. CSR fill: ev[] = {src, dinv[src]*dinv[dst]} packed int2
//   5. per layer: fp32 WMMA GEMM (h = X@W, W transposed+padded in LDS) then
//      CSR gather-sum aggregation (self-loop + bias + optional ReLU).
//      Feature matrices (51.2 MB) are L2-resident (192 MB L2) -> the gather
//      phase runs at L2 bandwidth with zero fp32 atomics.
// ---------------------------------------------------------------------------

typedef __attribute__((ext_vector_type(2))) float v2f;
typedef __attribute__((ext_vector_type(8))) float v8f;

#define HID 128
#define WPAD 132  // transposed-W LDS row stride; 132 % 64 == 4 -> conflict-free b64 reads

// -------------------------- CSR construction -------------------------------

__global__ void k_zero_i32(int* p, int n) {
  int i = blockIdx.x * 256 + threadIdx.x;
  if (i < n) p[i] = 0;
}

__global__ void k_count(const int* __restrict__ dst, int E, int* __restrict__ deg) {
  int e = blockIdx.x * 256 + threadIdx.x;
  if (e < E) atomicAdd(&deg[dst[e]], 1);
}

__global__ void k_dinv(const int* __restrict__ deg, float* __restrict__ dinv, int n) {
  int i = blockIdx.x * 256 + threadIdx.x;
  if (i < n) dinv[i] = rsqrtf((float)(deg[i] + 1));  // +1 self-loop, always > 0
}

// Block-local exclusive scan (256 elems / block), Hillis-Steele in LDS.
__global__ void k_scan1(const int* __restrict__ deg, int n,
                        int* __restrict__ row_ptr, int* __restrict__ bsum) {
  __shared__ int s[256];
  int t = threadIdx.x;
  int i = blockIdx.x * 256 + t;
  int v = (i < n) ? deg[i] : 0;
  s[t] = v;
  __syncthreads();
  for (int off = 1; off < 256; off <<= 1) {
    int add = (t >= off) ? s[t - off] : 0;
    __syncthreads();
    s[t] += add;
    __syncthreads();
  }
  int incl = s[t];
  if (i < n) row_ptr[i] = incl - v;       // exclusive
  if (t == 255) bsum[blockIdx.x] = incl;  // block total
}

// Serial scan of ~391 block sums; also writes row_ptr[n] = E. Setup-only cost.
__global__ void k_scan2(int* bsum, int nb, int* row_ptr, int n) {
  int run = 0;
  for (int b = 0; b < nb; ++b) { int t = bsum[b]; bsum[b] = run; run += t; }
  row_ptr[n] = run;
}

__global__ void k_scan3(int* __restrict__ row_ptr, const int* __restrict__ bsum,
                        int* __restrict__ cursor, int n) {
  int i = blockIdx.x * 256 + threadIdx.x;
  if (i < n) {
    int v = row_ptr[i] + bsum[blockIdx.x];
    row_ptr[i] = v;
    cursor[i]  = v;
  }
}

__global__ void k_fill(const int* __restrict__ src, const int* __restrict__ dst, int E,
                       const float* __restrict__ dinv, int* __restrict__ cursor,
                       int2* __restrict__ ev) {
  int e = blockIdx.x * 256 + threadIdx.x;
  if (e < E) {
    int s = src[e], d = dst[e];
    int pos = atomicAdd(&cursor[d], 1);
    ev[pos] = make_int2(s, __float_as_int(dinv[s] * dinv[d]));  // one b64 store
  }
}

// ------------------------------ WMMA GEMM ----------------------------------
// C[n x 128] = A[n x 128] @ W[128 x 128], fp32 via V_WMMA_F32_16X16X4_F32.
// Block: 256 threads = 8 waves; wave w owns rows [row0, row0+16) x all 128 cols.
// W is staged TRANSPOSED (sWt[c][k], stride 132) so each B fragment is a
// single contiguous ds_load_b64 with all 64 banks hit once per wave read.
__global__ __launch_bounds__(256) void k_gemm_wmma(const float* __restrict__ A,
                                                   const float* __restrict__ W,
                                                   float* __restrict__ C, int n) {
  __shared__ float sWt[HID * WPAD];  // 66 KB of the 320 KB/WGP pool
  for (int idx = threadIdx.x; idx < HID * HID; idx += 256) {
    int k = idx >> 7, c = idx & 127;
    sWt[c * WPAD + k] = W[idx];  // coalesced read, transposed write (one-time)
  }
  __syncthreads();

  const int lane = threadIdx.x & 31;
  const int wv   = threadIdx.x >> 5;  // 0..7
  const int r    = lane & 15;   // A: row-in-tile ; B/C: col-in-tile
  const int g    = lane >> 4;   // K half-select (A/B), M half-select (C)
  const int row0 = blockIdx.x * 128 + wv * 16;
  if (row0 >= n) return;  // wave-uniform

  // Clamp row for loads (keeps EXEC all-ones for WMMA); stores handled below.
  int rowA = row0 + r;
  if (rowA >= n) rowA = n - 1;
  const float* Arow = A + (size_t)rowA * HID;

  v8f acc[8] = {};  // 8 column tiles of 16 -> full 128-wide strip

#pragma unroll 1
  for (int k = 0; k < HID; k += 4) {
    // A frag: lane holds A[row][k+2g], A[row][k+2g+1]  (one global b64)
    const v2f a = *(const v2f*)(Arow + k + 2 * g);
#pragma unroll
    for (int t = 0; t < 8; ++t) {
      // B frag: lane holds W[k+2g][c0+r], W[k+2g+1][c0+r] == sWt[c0+r][k+2g..+1]
      const v2f b = *(const v2f*)(&sWt[(t * 16 + r) * WPAD + k + 2 * g]);
      acc[t] = __builtin_amdgcn_wmma_f32_16x16x4_f32(
          false, a, false, b, (short)0, acc[t], false, false);
    }
  }

  // C layout: VGPR i, lanes0-15 -> M=i, lanes16-31 -> M=i+8 ; N = c0 + r.
  float* Crow = C + (size_t)(row0 + 8 * g) * HID + r;
  if (row0 + 16 <= n) {
    // Full tile (always true when n % 16 == 0): unpredicated coalesced stores.
#pragma unroll
    for (int t = 0; t < 8; ++t)
#pragma unroll
      for (int i = 0; i < 8; ++i)
        Crow[(size_t)i * HID + t * 16] = acc[t][i];
  } else {
#pragma unroll
    for (int t = 0; t < 8; ++t)
#pragma unroll
      for (int i = 0; i < 8; ++i) {
        int row = row0 + i + 8 * g;
        if (row < n) C[(size_t)row * HID + t * 16 + r] = acc[t][i];
      }
  }
}

// --------------------------- CSR aggregation -------------------------------
// One wave32 per node; each lane owns a float4 of features (32*4 = 128).
// out[d] = sum_{e: dst=d} h[src_e]*w_e + h[d]*dinv[d]^2 + bias ; optional ReLU.
__global__ void k_aggregate(const float* __restrict__ h,
                            const int* __restrict__ row_ptr,
                            const int2* __restrict__ ev,
                            const float* __restrict__ dinv,
                            const float* __restrict__ bias,
                            float* __restrict__ out, int relu, int n) {
  const int lane = threadIdx.x & 31;
  const int wv   = threadIdx.x >> 5;
  const int node = blockIdx.x * 8 + wv;
  if (node >= n) return;

  const float4* hv = (const float4*)h;
  const float di = dinv[node];

  float4 acc = hv[(size_t)node * 32 + lane];  // self-loop term
  const float sw = di * di;
  acc.x *= sw; acc.y *= sw; acc.z *= sw; acc.w *= sw;

  const int beg = row_ptr[node];
  const int end = row_ptr[node + 1];
  for (int j = beg; j < end; ++j) {
    const int2  rec = ev[j];                  // {src, bits(norm)} : one b64 load
    const float w   = __int_as_float(rec.y);
    const float4 hs = hv[(size_t)rec.x * 32 + lane];  // 512B coalesced, L2-hot
    acc.x += hs.x * w; acc.y += hs.y * w; acc.z += hs.z * w; acc.w += hs.w * w;
  }

  const float4 bv = ((const float4*)bias)[lane];
  acc.x += bv.x; acc.y += bv.y; acc.z += bv.z; acc.w += bv.w;
  if (relu) {
    acc.x = fmaxf(acc.x, 0.f); acc.y = fmaxf(acc.y, 0.f);
    acc.z = fmaxf(acc.z, 0.f); acc.w = fmaxf(acc.w, 0.f);
  }
  ((float4*)out)[(size_t)node * 32 + lane] = acc;
}

// ------------------------------- launcher ----------------------------------

extern "C" void kernel_launch(void* const* d_in, const int* in_sizes, int n_in,
                              void* d_out, int out_size, void* d_ws, size_t ws_size,
                              hipStream_t stream) {
  const float* x     = (const float*)d_in[0];
  const int*   edges = (const int*)d_in[1];   // [2, E] flattened: src row, dst row
  const float* W1 = (const float*)d_in[2];
  const float* b1 = (const float*)d_in[3];
  const float* W2 = (const float*)d_in[4];
  const float* b2 = (const float*)d_in[5];
  const float* W3 = (const float*)d_in[6];
  const float* b3 = (const float*)d_in[7];

  const int N = in_sizes[0] / HID;   // 100000
  const int E = in_sizes[1] / 2;     // 1600000
  const int* src = edges;
  const int* dst = edges + E;

  // Workspace carve-out (256B aligned slices).
  char* p = (char*)d_ws;
  auto take = [&](size_t bytes) -> void* {
    void* r = (void*)p;
    p += (bytes + 255) & ~(size_t)255;
    return r;
  };
  int*   deg     = (int*)  take((size_t)N * 4);
  int*   row_ptr = (int*)  take((size_t)(N + 1) * 4);
  int*   cursor  = (int*)  take((size_t)N * 4);
  int*   bsum    = (int*)  take((size_t)1024 * 4);
  int2*  ev      = (int2*) take((size_t)E * 8);
  float* dinv    = (float*)take((size_t)N * 4);
  float* bufA    = (float*)take((size_t)N * HID * 4);
  float* bufB    = (float*)take((size_t)N * HID * 4);
  (void)ws_size; (void)n_in; (void)out_size;

  const int nbN = (N + 255) / 256;
  const int nbE = (E + 255) / 256;

  // --- build normalization + CSR (by destination) ---
  k_zero_i32<<<nbN, 256, 0, stream>>>(deg, N);
  k_count<<<nbE, 256, 0, stream>>>(dst, E, deg);
  k_dinv<<<nbN, 256, 0, stream>>>(deg, dinv, N);
  k_scan1<<<nbN, 256, 0, stream>>>(deg, N, row_ptr, bsum);
  k_scan2<<<1, 1, 0, stream>>>(bsum, nbN, row_ptr, N);
  k_scan3<<<nbN, 256, 0, stream>>>(row_ptr, bsum, cursor, N);
  k_fill<<<nbE, 256, 0, stream>>>(src, dst, E, dinv, cursor, ev);

  const int gGemm = (N + 127) / 128;  // 8 waves/block, 16 rows/wave
  const int gAgg  = (N + 7) / 8;      // 8 nodes/block (1 wave each)

  // --- layer 1 ---
  k_gemm_wmma<<<gGemm, 256, 0, stream>>>(x, W1, bufB, N);
  k_aggregate<<<gAgg, 256, 0, stream>>>(bufB, row_ptr, ev, dinv, b1, bufA, 1, N);
  // --- layer 2 ---
  k_gemm_wmma<<<gGemm, 256, 0, stream>>>(bufA, W2, bufB, N);
  k_aggregate<<<gAgg, 256, 0, stream>>>(bufB, row_ptr, ev, dinv, b2, bufA, 1, N);
  // --- layer 3 (no ReLU, straight to d_out) ---
  k_gemm_wmma<<<gGemm, 256, 0, stream>>>(bufA, W3, bufB, N);
  k_aggregate<<<gAgg, 256, 0, stream>>>(bufB, row_ptr, ev, dinv, b3,
                                        (float*)d_out, 0, N);
}